// CiLayer_28226525069548
// MI455X (gfx1250) — compile-verified
//
#include <hip/hip_runtime.h>
#include <math.h>

// 180 / pi
#define RAD2DEG 57.295779513082320877f

typedef float v4f __attribute__((ext_vector_type(4)));
typedef float v2f __attribute__((ext_vector_type(2)));

// signed_acos(x, y) * DEG, with t = x / sqrt(x^2+y^2) computed as
// x * rsqrt(x^2+y^2) via v_rsq_f32 + one Newton-Raphson step (~1 ulp on t,
// far fewer VALU ops than the IEEE sqrt + div_scale/div_fmas sequences).
__device__ __forceinline__ float signed_acos_deg(float x, float y) {
    float s = __builtin_fmaf(x, x, y * y);
    float r = __builtin_amdgcn_rsqf(s);              // v_rsq_f32
    // Newton-Raphson: r <- r + 0.5*r*(1 - s*r*r)
    float e = __builtin_fmaf(-s * r, r, 1.0f);
    r = __builtin_fmaf(0.5f * r, e, r);
    // s==0: rsq -> inf, NR -> NaN, t -> NaN, acos -> NaN (matches ref 0/0).
    float t = x * r;
    float a = acosf(t);                              // NaN propagates
    a = (y < 0.0f) ? -a : a;                         // false for NaN(y) (jnp.where)
    return a * RAD2DEG;
}

__device__ __forceinline__ v2f angles_of(v4f v) {
    v2f r;
    r.x = signed_acos_deg(v.x, v.y);
    r.y = signed_acos_deg(v.z, v.w);
    return r;
}

// Hot kernel: n guaranteed (by the launcher) to be a multiple of 512
// positions = 2 * 256 threads. Zero bounds checks -> zero EXEC-mask
// manipulation: straight-line  2x global_load_b128 NT (clause) -> VALU ->
// 2x global_store_b64 NT. Wave-strided pair keeps both loads unit-stride
// coalesced; two independent chains hide TRANS (rsq/acos) latency.
__global__ __launch_bounds__(256) void signed_acos_angles_exact(
        const v4f* __restrict__ in, v2f* __restrict__ out) {
    const int i0 = blockIdx.x * 512 + threadIdx.x;
    const int i1 = i0 + 256;

    v4f v0 = __builtin_nontemporal_load(&in[i0]);
    v4f v1 = __builtin_nontemporal_load(&in[i1]);
    v2f r0 = angles_of(v0);
    v2f r1 = angles_of(v1);
    __builtin_nontemporal_store(r0, &out[i0]);
    __builtin_nontemporal_store(r1, &out[i1]);
}

// Generic fallback for arbitrary n (one position per thread, guarded).
__global__ __launch_bounds__(256) void signed_acos_angles_guarded(
        const v4f* __restrict__ in, v2f* __restrict__ out, int n) {
    const int i = blockIdx.x * blockDim.x + threadIdx.x;
    if (i >= n) return;
    v4f v = __builtin_nontemporal_load(&in[i]);
    v2f r = angles_of(v);
    __builtin_nontemporal_store(r, &out[i]);
}

extern "C" void kernel_launch(void* const* d_in, const int* in_sizes, int n_in,
                              void* d_out, int out_size, void* d_ws, size_t ws_size,
                              hipStream_t stream) {
    const v4f* in = (const v4f*)d_in[0];
    v2f* out = (v2f*)d_out;

    // in_sizes[0] is the flat float count of the [B, L, 4] input.
    const int n = in_sizes[0] / 4;   // number of (x,y) pair positions

    if (n % 512 == 0) {
        // Reference shape: 512*16384 = 8388608 positions -> 16384 blocks.
        signed_acos_angles_exact<<<n / 512, 256, 0, stream>>>(in, out);
    } else {
        const int threads = 256;
        const int blocks = (n + threads - 1) / threads;
        signed_acos_angles_guarded<<<blocks, threads, 0, stream>>>(in, out, n);
    }
}